// real_KW3_52699248721958
// MI455X (gfx1250) — compile-verified
//
#include <hip/hip_runtime.h>
#include <cstdint>

// real_KW3: per-row (C=1000) softmax(T=20) of o1,o2,o3 and their mean, then
// mean of the three L1 distances to the mimic distribution. Memory-bound:
// ~197MB read -> ~8.4us floor @ 23.3TB/s. One block per row; CDNA5 async
// global->LDS staging; exps cached in LDS so each transcendental is computed
// exactly once.

#define C_DIM   1000
#define TPB     256
#define NWAVES  (TPB / 32)

// Low 32 bits of a generic pointer to LDS are the wave-relative LDS byte
// address (CDNA5 flat-aperture rule: upper bits are the shared aperture).
__device__ __forceinline__ uint32_t lds_addr32(const void* p) {
    return (uint32_t)(uintptr_t)p;
}

// CDNA5 async DMA: 16 bytes global -> LDS, tracked by ASYNCcnt (no VGPR data).
__device__ __forceinline__ void async_g2l_b128(uint32_t lds, const void* gptr) {
    asm volatile("global_load_async_to_lds_b128 %0, %1, off"
                 :: "v"(lds), "v"(gptr)
                 : "memory");
}

__device__ __forceinline__ void wait_async_all() {
    asm volatile("s_wait_asynccnt 0" ::: "memory");
}

__device__ __forceinline__ float wave_sum(float v) {
#pragma unroll
    for (int m = 16; m >= 1; m >>= 1)
        v += __shfl_xor(v, m, 32);   // wave32 on gfx1250
    return v;
}

__global__ void __launch_bounds__(TPB)
kw3_kernel(const float* __restrict__ o1,
           const float* __restrict__ o2,
           const float* __restrict__ o3,
           float* __restrict__ out,
           int n_rows)
{
    __shared__ __align__(16) float sA[C_DIM];   // exp cache for o1
    __shared__ __align__(16) float sB[C_DIM];   // exp cache for o2
    __shared__ __align__(16) float sC[C_DIM];   // exp cache for o3
    __shared__ __align__(16) float sM[C_DIM];   // exp cache for mimic
    __shared__ float sred[NWAVES][4];
    __shared__ float sinv[4];

    const int row = blockIdx.x;
    if (row >= n_rows) return;
    const int tid  = threadIdx.x;
    const int lane = tid & 31;
    const int wave = tid >> 5;
    const size_t base = (size_t)row * C_DIM;

    // ---- Stage the three rows into LDS: 4000B/row = 250 B128 async DMAs ----
    if (tid < (C_DIM * 4) / 16) {
        const int byteoff = tid * 16;
        async_g2l_b128(lds_addr32((const char*)sA + byteoff),
                       (const char*)(o1 + base) + byteoff);
        async_g2l_b128(lds_addr32((const char*)sB + byteoff),
                       (const char*)(o2 + base) + byteoff);
        async_g2l_b128(lds_addr32((const char*)sC + byteoff),
                       (const char*)(o3 + base) + byteoff);
    }
    wait_async_all();      // per-wave ASYNCcnt drain
    __syncthreads();       // all waves' DMAs visible in LDS

    // ---- Pass 1: exps (computed once, cached in LDS) + 4 row sums ----
    const float invT = 1.0f / 20.0f;
    float sa = 0.0f, sb = 0.0f, sc = 0.0f, sm = 0.0f;
#pragma unroll 4
    for (int i = tid; i < C_DIM; i += TPB) {
        const float a = sA[i], b = sB[i], c = sC[i];
        const float m = (a + b + c) * (1.0f / 3.0f);
        const float ea = __expf(a * invT);
        const float eb = __expf(b * invT);
        const float ec = __expf(c * invT);
        const float em = __expf(m * invT);
        sA[i] = ea; sB[i] = eb; sC[i] = ec; sM[i] = em;  // owner-exclusive slots
        sa += ea; sb += eb; sc += ec; sm += em;
    }

    sa = wave_sum(sa); sb = wave_sum(sb); sc = wave_sum(sc); sm = wave_sum(sm);
    if (lane == 0) {
        sred[wave][0] = sa; sred[wave][1] = sb;
        sred[wave][2] = sc; sred[wave][3] = sm;
    }
    __syncthreads();
    if (tid < 4) {
        float t = 0.0f;
#pragma unroll
        for (int w = 0; w < NWAVES; ++w) t += sred[w][tid];
        sinv[tid] = 1.0f / t;
    }
    __syncthreads();
    const float ra = sinv[0], rb = sinv[1], rc = sinv[2], rm = sinv[3];

    // ---- Pass 2: L1 distances from cached exps ----
    float acc = 0.0f;
#pragma unroll 4
    for (int i = tid; i < C_DIM; i += TPB) {
        const float pm = sM[i] * rm;
        acc += fabsf(sA[i] * ra - pm)
             + fabsf(sB[i] * rb - pm)
             + fabsf(sC[i] * rc - pm);
    }
    acc = wave_sum(acc);
    __syncthreads();               // sred reuse hazard
    if (lane == 0) sred[wave][0] = acc;
    __syncthreads();
    if (tid == 0) {
        float t = 0.0f;
#pragma unroll
        for (int w = 0; w < NWAVES; ++w) t += sred[w][0];
        out[row] = t * (1.0f / 3.0f);
    }
}

extern "C" void kernel_launch(void* const* d_in, const int* in_sizes, int n_in,
                              void* d_out, int out_size, void* d_ws, size_t ws_size,
                              hipStream_t stream) {
    (void)d_ws; (void)ws_size; (void)out_size;
    const float* o1 = (const float*)d_in[0];
    const float* o2 = (const float*)d_in[1];
    const float* o3 = (const float*)d_in[2];

    const int prefix = (n_in > 3) ? in_sizes[3] : 0;   // real_KW prefix (0 here)
    float* out = (float*)d_out;
    if (prefix > 0) {
        hipMemcpyAsync(out, d_in[3], (size_t)prefix * sizeof(float),
                       hipMemcpyDeviceToDevice, stream);
    }

    const int n_rows = in_sizes[0] / C_DIM;            // 16384
    kw3_kernel<<<n_rows, TPB, 0, stream>>>(o1, o2, o3, out + prefix, n_rows);
}